// QuantumLikeLayer_69672959476140
// MI455X (gfx1250) — compile-verified
//
#include <hip/hip_runtime.h>

// CDNA5 / gfx1250 FWHT via V_WMMA_F32_16X16X4_F32.
// One wave32 processes one row of 2048 f32.
//   n = a*128 + b*8 + c, a,b in [0,16), c in [0,8)
//   H2048 = H16(a) o H16(b) o H8(c)
// Two H16 stages are WMMA f32 16x16 GEMMs (K=16 as 4 chained 16x16x4 WMMAs),
// H8 stage is 3 register butterflies across c-tiles.

typedef __attribute__((ext_vector_type(2))) float v2f;
typedef __attribute__((ext_vector_type(4))) float f4;
typedef __attribute__((ext_vector_type(8))) float v8f;

#define DIM    2048
#define WAVES  8
#define NTHR   (WAVES * 32)

__device__ __forceinline__ float hsign16(int m, int k, float s) {
    // H16[m][k] = (-1)^popcount(m&k), scaled by s
    return (__popc(m & k) & 1) ? -s : s;
}

__global__ __launch_bounds__(NTHR) void fwht2048_wmma_kernel(
    const float* __restrict__ x, float* __restrict__ out, int rows) {
    __shared__ float lds[WAVES][DIM];

    const int lane = threadIdx.x & 31;
    const int wid  = threadIdx.x >> 5;
    const int row  = blockIdx.x * WAVES + wid;
    if (row >= rows) return;  // never taken for 16384 rows; keeps EXEC full

    const float* __restrict__ xr   = x   + (size_t)row * DIM;
    float* __restrict__       orow = out + (size_t)row * DIM;
    float* buf = lds[wid];

    // ---- Stage in: coalesced global -> LDS (row in natural layout) ----
    #pragma unroll
    for (int i = 0; i < 16; ++i) {
        f4 v = ((const f4*)xr)[i * 32 + lane];
        ((f4*)buf)[i * 32 + lane] = v;
    }

    const int m  = lane & 15;   // A-operand M row / B-operand N column
    const int hi = lane >> 4;   // lane-half selects K sub-pair per WMMA layout

    // Constant A operand: H16 scaled by 0.25 (folds 4 of 11 1/sqrt2 factors),
    // 4 chunks of K=4.  A layout (16x4 f32): vgpr0 holds K=k0 (lanes 0-15) and
    // K=k0+2 (lanes 16-31); vgpr1 holds K=k0+1 / K=k0+3.
    v2f A[4];
    #pragma unroll
    for (int j = 0; j < 4; ++j) {
        int k = 4 * j + 2 * hi;
        A[j].x = hsign16(m, k,     0.25f);
        A[j].y = hsign16(m, k + 1, 0.25f);
    }

    // ---- Stage 1: U(:,:,c) = H16 * X(:,:,c)  (sum over a') ----
    // B[K=a'][N=b]: lane holds N=b=m, K per chunk/half. Results written back
    // to LDS in natural (a,b,c) layout (in-order DS within a wave => safe).
    #pragma unroll
    for (int c = 0; c < 8; ++c) {
        v8f acc = {};
        #pragma unroll
        for (int j = 0; j < 4; ++j) {
            int a0 = 4 * j + 2 * hi;
            v2f B;
            B.x = buf[(a0    ) * 128 + m * 8 + c];
            B.y = buf[(a0 + 1) * 128 + m * 8 + c];
            acc = __builtin_amdgcn_wmma_f32_16x16x4_f32(
                false, A[j], false, B, (short)0, acc, false, false);
        }
        // D layout: vgpr v -> M=a=v+8*hi, N=b=m
        #pragma unroll
        for (int v = 0; v < 8; ++v)
            buf[(v + 8 * hi) * 128 + m * 8 + c] = acc[v];
    }

    // ---- Stage 2: W = H16 * U^T  (== (U * H16)^T, H16 symmetric) ----
    // B[K=b'][N=a]: lane holds N=a=m; read from natural layout. Tiles kept in
    // registers (8 x v8f = 64 VGPRs).
    v8f w[8];
    #pragma unroll
    for (int c = 0; c < 8; ++c) {
        v8f acc = {};
        #pragma unroll
        for (int j = 0; j < 4; ++j) {
            int b0 = 4 * j + 2 * hi;
            v2f B;
            B.x = buf[m * 128 + (b0    ) * 8 + c];
            B.y = buf[m * 128 + (b0 + 1) * 8 + c];
            acc = __builtin_amdgcn_wmma_f32_16x16x4_f32(
                false, A[j], false, B, (short)0, acc, false, false);
        }
        w[c] = acc;  // element (b = v+8*hi, a = m) of V[a][b] (x16 scaled)
    }

    // ---- Stage 3: H8 across c (3 butterfly stages, pure VALU) ----
    #pragma unroll
    for (int h = 1; h < 8; h <<= 1) {
        #pragma unroll
        for (int g = 0; g < 8; g += 2 * h) {
            #pragma unroll
            for (int t = 0; t < h; ++t) {
                v8f p = w[g + t], q = w[g + t + h];
                w[g + t]     = p + q;
                w[g + t + h] = p - q;
            }
        }
    }

    // Sign flip for n >= 3N/4  <=>  a >= 12; a == m for every element this
    // lane owns, so the sign is per-lane uniform. Fold remaining 2^-1.5 norm.
    const float mult = (m >= 12) ? -0.35355339059327373f : 0.35355339059327373f;

    // Scatter results back to LDS in natural layout, then stream out coalesced.
    #pragma unroll
    for (int c = 0; c < 8; ++c) {
        #pragma unroll
        for (int v = 0; v < 8; ++v)
            buf[m * 128 + (v + 8 * hi) * 8 + c] = w[c][v] * mult;
    }

    #pragma unroll
    for (int i = 0; i < 16; ++i)
        ((f4*)orow)[i * 32 + lane] = ((const f4*)buf)[i * 32 + lane];
}

extern "C" void kernel_launch(void* const* d_in, const int* in_sizes, int n_in,
                              void* d_out, int out_size, void* d_ws, size_t ws_size,
                              hipStream_t stream) {
    (void)n_in; (void)out_size; (void)d_ws; (void)ws_size;
    const float* x = (const float*)d_in[0];
    float* out = (float*)d_out;
    const int rows = in_sizes[0] / DIM;            // 16384
    const int blocks = (rows + WAVES - 1) / WAVES; // 2048
    fwht2048_wmma_kernel<<<blocks, NTHR, 0, stream>>>(x, out, rows);
}